// GraphDilatedKNN_45612552683641
// MI455X (gfx1250) — compile-verified
//
#include <hip/hip_runtime.h>

typedef __attribute__((ext_vector_type(2))) float v2f;
typedef __attribute__((ext_vector_type(8))) float v8f;

#define NS   20            // SAMPLE_NUM
#define NDIL 16            // len(DILATED_IDX)
#define WPB  8             // waves per block (256 threads, wave32)

// Kernel 1: WMMA-based ball query -> dilated neighbor ids.
// One wave handles a 16-row query tile of one batch, scanning candidate
// points in 16-column tiles with V_WMMA_F32_16X16X4_F32 (K=3 padded to 4).
__global__ __launch_bounds__(256) void GraphDilatedKNN_ballquery_wmma(
    const float* __restrict__ xyz, int* __restrict__ ids, int B, int N) {
  const int lane = threadIdx.x & 31;
  const int widx = threadIdx.x >> 5;
  const int gwave = blockIdx.x * WPB + widx;
  const int tilesPerBatch = N >> 4;
  const int b = gwave / tilesPerBatch;
  const int tile = gwave - b * tilesPerBatch;
  if (b >= B) return;                       // wave-uniform
  const int m = lane & 15;
  const int i0 = tile << 4;
  const float* xb = xyz + (size_t)b * N * 3;

  // Query point for row m (both half-waves load the same 16 rows).
  float qx = xb[(i0 + m) * 3 + 0];
  float qy = xb[(i0 + m) * 3 + 1];
  float qz = xb[(i0 + m) * 3 + 2];
  float rn = qx * qx + qy * qy + qz * qz;

  // A fragment (16x4 f32): lane<16 -> (K0,K1)=(x,y); lane>=16 -> (K2,K3)=(z,1)
  v2f a;
  a.x = (lane < 16) ? qx : qz;
  a.y = (lane < 16) ? qy : 1.0f;

  // C fragment: c[v] = ||q_M||^2 with M = v + 8*(lane>=16)
  v8f c;
#pragma unroll
  for (int v = 0; v < 8; ++v) {
    int M = v + ((lane >= 16) ? 8 : 0);
    c[v] = __shfl(rn, M, 32);
  }

  __shared__ int sel_s[WPB][16][NS];
  int* sel = &sel_s[widx][m][0];            // lane r (<16) owns row r
  int cnt = (lane < 16) ? 0 : NS;           // upper lanes pre-"done" for votes
  const float r2 = 256.0f;                  // RADIUS^2

  for (int j0 = 0; j0 < N; j0 += 16) {
    // Candidate point for column m of this tile.
    float px = xb[(j0 + m) * 3 + 0];
    float py = xb[(j0 + m) * 3 + 1];
    float pz = xb[(j0 + m) * 3 + 2];
    float pn = px * px + py * py + pz * pz;

    // B fragment (4x16 f32): col = (-2x, -2y, -2z, ||p||^2)
    v2f bf;
    bf.x = (lane < 16) ? (-2.0f * px) : (-2.0f * pz);
    bf.y = (lane < 16) ? (-2.0f * py) : pn;

    // D = A*B + C = ||q||^2 - 2 q.p + ||p||^2  (16x16 tile of squared dist)
    v8f d = __builtin_amdgcn_wmma_f32_16x16x4_f32(
        false, a, false, bf, (short)0, c, false, false);

    // One ballot per D VGPR: bits[15:0] = row v hits, bits[31:16] = row v+8.
    unsigned msk[8];
#pragma unroll
    for (int v = 0; v < 8; ++v)
      msk[v] = __builtin_amdgcn_ballot_w32(d[v] < r2);

    if (lane < 16) {
      int vs = lane & 7;
      unsigned bm = msk[0];
      bm = (vs == 1) ? msk[1] : bm;
      bm = (vs == 2) ? msk[2] : bm;
      bm = (vs == 3) ? msk[3] : bm;
      bm = (vs == 4) ? msk[4] : bm;
      bm = (vs == 5) ? msk[5] : bm;
      bm = (vs == 6) ? msk[6] : bm;
      bm = (vs == 7) ? msk[7] : bm;
      if (lane & 8) bm >>= 16;
      bm &= 0xFFFFu;
      while (bm != 0u && cnt < NS) {       // ascending j == top_k(-idx) order
        int bit = __builtin_ctz(bm);
        bm &= bm - 1u;
        sel[cnt++] = j0 + bit;
      }
    }
    // Uniform early exit once every row has 20 neighbors.
    if (__builtin_amdgcn_ballot_w32(cnt >= NS) == 0xFFFFFFFFu) break;
  }

  if (lane < 16) {
    int first = (cnt > 0) ? sel[0] : (i0 + m);   // self always in-radius
    for (int k = cnt; k < NS; ++k) sel[k] = first;
    const int DIL[NDIL] = {1, 2, 3, 4, 6, 7, 8, 9,
                           11, 12, 13, 14, 16, 17, 18, 19};
    int i = i0 + m;
    int* idrow = ids + ((size_t)b * N + i) * NDIL;
#pragma unroll
    for (int k = 0; k < NDIL; ++k) idrow[k] = sel[DIL[k]];
  }
}

// Kernel 2: bandwidth phase. One 256-thread block per (b,i); gathers xyz
// (3 ch) and feature (128 ch) at the 16 dilated ids; writes are contiguous
// over the trailing (i,k) dims of the (B,C,N,S) outputs.
__global__ __launch_bounds__(256) void GraphDilatedKNN_gather(
    const float* __restrict__ xyz, const float* __restrict__ feat,
    const int* __restrict__ ids, float* __restrict__ out,
    int B, int N, int C) {
  const int bi = blockIdx.x;                // 0 .. B*N-1
  const int b = bi / N;
  const int i = bi - b * N;
  const int t = threadIdx.x;

  __shared__ int sid[NDIL];
  if (t < NDIL) sid[t] = ids[(size_t)bi * NDIL + t];
  __syncthreads();

  const int k = t & 15;
  const int id = sid[k];

  // dilated_xyz section: (B, 3, N, 16)
  if (t < 48) {
    int dax = t >> 4;                        // 0..2
    out[(((size_t)b * 3 + dax) * N + i) * NDIL + k] =
        xyz[((size_t)b * N + id) * 3 + dax];
  }

  // dilated_feature section: (B, C, N, 16), after the xyz section
  float* out_feat = out + (size_t)B * 3 * N * NDIL;
  const float* fb = feat + ((size_t)b * N + id) * C;
  for (int cc = (t >> 4); cc < C; cc += 16) {
    out_feat[(((size_t)b * C + cc) * N + i) * NDIL + k] = fb[cc];
  }
}

extern "C" void kernel_launch(void* const* d_in, const int* in_sizes, int n_in,
                              void* d_out, int out_size, void* d_ws, size_t ws_size,
                              hipStream_t stream) {
  (void)in_sizes; (void)n_in; (void)out_size; (void)ws_size;
  const float* xyz  = (const float*)d_in[0];   // (8, 4096, 3) f32
  const float* feat = (const float*)d_in[1];   // (8, 4096, 128) f32
  float* out = (float*)d_out;                  // xyz(8,3,4096,16) ++ feat(8,128,4096,16)
  int* ids = (int*)d_ws;                       // (8, 4096, 16) i32 scratch (2 MiB)

  constexpr int B = 8, N = 4096, C = 128;

  const int totalWaves = B * (N / 16);                   // 2048
  dim3 blk1(32 * WPB);
  dim3 grd1(totalWaves / WPB);                           // 256 blocks
  GraphDilatedKNN_ballquery_wmma<<<grd1, blk1, 0, stream>>>(xyz, ids, B, N);

  dim3 blk2(256);
  dim3 grd2(B * N);                                      // 32768 blocks
  GraphDilatedKNN_gather<<<grd2, blk2, 0, stream>>>(xyz, feat, ids, out, B, N, C);
}